// Unpooling2D_2293512536994
// MI455X (gfx1250) — compile-verified
//
#include <hip/hip_runtime.h>

// SegNet 2x2 max-unpooling, channels-last, stride 2, 'valid'.
//   pool_input:        [B=16, H=128, W=128, C=128] f32
//   recreated_output:  [B=16, 64, 64, 128] f32
//   out:               [B=16, 128, 128, 128] f32
//
// Pure streaming scatter (~302 MB moved, ~0 FLOPs) -> HBM-bound (~13us @ 23.3 TB/s).
// Strategy: one float4 (4 channels) per lane; a wave32 covers exactly one
// window position's 128 channels, so every b128 global op is one contiguous,
// fully coalesced 512B transaction. Non-temporal hints keep this single-use
// stream from polluting L2.

typedef float v4f __attribute__((ext_vector_type(4)));

__global__ __launch_bounds__(256) void unpool2x2_argmax_scatter(
    const float* __restrict__ pin,   // [B,128,128,128]
    const float* __restrict__ rec,   // [B, 64, 64,128]
    float* __restrict__ out,         // [B,128,128,128]
    int n_tasks)                     // B*64*64*(C/4)
{
    constexpr int C  = 128;          // channels
    constexpr int W  = 128;          // input width
    constexpr int CV = C / 4;        // float4 chunks per channel row (32)
    constexpr long ROW = (long)W * C;    // stride between input rows (floats)

    int t = blockIdx.x * blockDim.x + threadIdx.x;
    if (t >= n_tasks) return;

    // t -> (b, h_, w_, cv); all dims are powers of two.
    int cv = t & (CV - 1);           // 0..31
    int w_ = (t >> 5)  & 63;         // 0..63
    int h_ = (t >> 11) & 63;         // 0..63
    int b  = t >> 17;                // 0..15

    // Base of the 2x2 window in the full-resolution tensor.
    long base  = ((((long)b * 128 + 2 * h_) * W) + 2 * w_) * C + cv * 4;
    long rbase = ((((long)b * 64 + h_) * 64) + w_) * C + cv * 4;

    // Window slot order must match the reference reshape: slot = py*2 + px.
    //   slot0 = (2h,   2w)    slot1 = (2h,   2w+1)
    //   slot2 = (2h+1, 2w)    slot3 = (2h+1, 2w+1)
    v4f v0 = __builtin_nontemporal_load((const v4f*)(pin + base));
    v4f v1 = __builtin_nontemporal_load((const v4f*)(pin + base + C));
    v4f v2 = __builtin_nontemporal_load((const v4f*)(pin + base + ROW));
    v4f v3 = __builtin_nontemporal_load((const v4f*)(pin + base + ROW + C));
    v4f r  = __builtin_nontemporal_load((const v4f*)(rec + rbase));

    v4f o0 = {0.f, 0.f, 0.f, 0.f};
    v4f o1 = {0.f, 0.f, 0.f, 0.f};
    v4f o2 = {0.f, 0.f, 0.f, 0.f};
    v4f o3 = {0.f, 0.f, 0.f, 0.f};

#pragma unroll
    for (int j = 0; j < 4; ++j) {
        // First-occurrence argmax (strict '>' so ties keep the earliest slot),
        // matching jnp.argmax over the window axis.
        float m = v0[j];
        int idx = 0;
        if (v1[j] > m) { m = v1[j]; idx = 1; }
        if (v2[j] > m) { m = v2[j]; idx = 2; }
        if (v3[j] > m) { m = v3[j]; idx = 3; }
        float rv = r[j];
        o0[j] = (idx == 0) ? rv : 0.0f;
        o1[j] = (idx == 1) ? rv : 0.0f;
        o2[j] = (idx == 2) ? rv : 0.0f;
        o3[j] = (idx == 3) ? rv : 0.0f;
    }

    __builtin_nontemporal_store(o0, (v4f*)(out + base));
    __builtin_nontemporal_store(o1, (v4f*)(out + base + C));
    __builtin_nontemporal_store(o2, (v4f*)(out + base + ROW));
    __builtin_nontemporal_store(o3, (v4f*)(out + base + ROW + C));
}

extern "C" void kernel_launch(void* const* d_in, const int* in_sizes, int n_in,
                              void* d_out, int out_size, void* d_ws, size_t ws_size,
                              hipStream_t stream) {
    (void)n_in; (void)out_size; (void)d_ws; (void)ws_size;
    const float* pin = (const float*)d_in[0];   // pool_input       [16,128,128,128]
    const float* rec = (const float*)d_in[1];   // recreated_output [16, 64, 64,128]
    float* out = (float*)d_out;                 // output           [16,128,128,128]

    // One float4-of-channels per thread: B*h*w*(C/4) = in_sizes[0] / 16.
    int n_tasks = in_sizes[0] / 16;             // 2,097,152
    const int block = 256;                      // 8 wave32s
    const int grid  = (n_tasks + block - 1) / block;  // 8192

    unpool2x2_argmax_scatter<<<grid, block, 0, stream>>>(pin, rec, out, n_tasks);
}